// SingleProp_75935021793752
// MI455X (gfx1250) — compile-verified
//
#include <hip/hip_runtime.h>
#include <hip/hip_bf16.h>

#define EMBED     128
#define HIDDEN    128
#define NUM_RELS  8
#define N_GRAPHS  64

typedef __attribute__((ext_vector_type(2))) float v2f;
typedef __attribute__((ext_vector_type(4))) float v4f;
typedef __attribute__((ext_vector_type(8))) float v8f;

// ---------------------------------------------------------------- embedding
__global__ void embed_kernel(const int* __restrict__ x, const float* __restrict__ emb,
                             float* __restrict__ feat, int n) {
    int i = blockIdx.x * blockDim.x + threadIdx.x;   // one thread per (node, 4 cols)
    int node = i >> 5;
    int c4   = (i & 31) << 2;
    if (node >= n) return;
    int row = x[node];
    *(v4f*)(feat + (size_t)node * EMBED + c4) = *(const v4f*)(emb + (size_t)row * EMBED + c4);
}

// ---------------------------------------------------------------- degree count
__global__ void count_kernel(const int* __restrict__ dst, const int* __restrict__ et,
                             int* __restrict__ cnt, int e) {
    int i = blockIdx.x * blockDim.x + threadIdx.x;
    if (i >= e) return;
    atomicAdd(&cnt[(size_t)dst[i] * NUM_RELS + et[i]], 1);
}

// ---------------------------------------------------------------- per-relation scatter-add
// one wave (32 lanes) per edge, lane handles 4 floats
__global__ void scatter_kernel(const int* __restrict__ src, const int* __restrict__ dst,
                               const int* __restrict__ et, const float* __restrict__ feat,
                               float* __restrict__ tmp, int e, int r) {
    int lane = threadIdx.x & 31;
    int eid  = blockIdx.x * (blockDim.x >> 5) + (threadIdx.x >> 5);
    if (eid >= e) return;
    if (et[eid] != r) return;
    int s = src[eid], d = dst[eid];
    v4f v = *(const v4f*)(feat + (size_t)s * EMBED + lane * 4);
    float* td = tmp + (size_t)d * EMBED + lane * 4;
    atomicAdd(td + 0, v.x);
    atomicAdd(td + 1, v.y);
    atomicAdd(td + 2, v.z);
    atomicAdd(td + 3, v.w);
}

// ---------------------------------------------------------------- WMMA GEMM
// out[n, 0:128] (+)= scale(n) * A[n, 0:128] @ W[128,128]
// mode 0: overwrite; mode 1: accumulate into out; mode 2: accumulate + bias + relu
// scale(n) = 1 / max(cnt[n*NUM_RELS + rel], 1) when cnt != nullptr, else 1
__global__ __launch_bounds__(256)
void gemm128_wmma(const float* __restrict__ A, const float* __restrict__ W,
                  float* __restrict__ out, const int* __restrict__ cnt, int rel,
                  const float* __restrict__ bias, int mode, int n) {
    __shared__ float ldsW[EMBED * HIDDEN];           // 64 KB: whole weight matrix
    for (int i = threadIdx.x * 4; i < EMBED * HIDDEN; i += blockDim.x * 4)
        *(v4f*)(ldsW + i) = *(const v4f*)(W + i);
    __syncthreads();

    const int wave = threadIdx.x >> 5;
    const int lane = threadIdx.x & 31;
    const int row0 = (blockIdx.x * 8 + wave) * 16;   // 16 rows per wave
    if (row0 >= n) return;                           // N is a multiple of 16: full groups only

    const int m    = lane & 15;                      // A: M index / B,C: N index within tile
    const int half = lane >> 4;                      // 0 or 1
    const int kh   = half * 2;                       // K sub-offset for A/B fragments

    float s = 1.0f;
    if (cnt)
        s = 1.0f / fmaxf((float)cnt[(size_t)(row0 + m) * NUM_RELS + rel], 1.0f);

    // accumulators: 8 tiles of 16x16 covering N = 0..127
    v8f acc[8];
    if (mode == 0) {
        v8f z = {};
        #pragma unroll
        for (int t = 0; t < 8; ++t) acc[t] = z;
    } else {
        #pragma unroll
        for (int t = 0; t < 8; ++t) {
            v8f c;
            #pragma unroll
            for (int v = 0; v < 8; ++v)
                c[v] = out[(size_t)(row0 + half * 8 + v) * HIDDEN + t * 16 + m];
            acc[t] = c;
        }
    }

    const float* Arow = A + (size_t)row0 * EMBED;
    #pragma unroll 2
    for (int k0 = 0; k0 < EMBED; k0 += 4) {
        // A fragment: 16x4, lane L holds A[m][k0+kh .. k0+kh+1], scaled by 1/deg
        v2f a = *(const v2f*)(Arow + (size_t)m * EMBED + k0 + kh);
        a.x *= s; a.y *= s;
        #pragma unroll
        for (int t = 0; t < 8; ++t) {
            // B fragment: 4x16, lane L holds W[k0+kh .. +1][t*16+m]
            v2f b;
            b.x = ldsW[(k0 + kh) * HIDDEN + t * 16 + m];
            b.y = ldsW[(k0 + kh + 1) * HIDDEN + t * 16 + m];
            acc[t] = __builtin_amdgcn_wmma_f32_16x16x4_f32(
                false, a, false, b, (short)0, acc[t], false, false);
        }
    }

    #pragma unroll
    for (int t = 0; t < 8; ++t) {
        #pragma unroll
        for (int v = 0; v < 8; ++v) {
            int row = row0 + half * 8 + v;
            int col = t * 16 + m;
            float val = acc[t][v];
            if (mode == 2) val = fmaxf(val + bias[col], 0.0f);
            out[(size_t)row * HIDDEN + col] = val;
        }
    }
}

// ---------------------------------------------------------------- masked segment mean (sum part)
__global__ void pool_kernel(const float* __restrict__ h, const int* __restrict__ batch,
                            const int* __restrict__ ntype, float* __restrict__ seg_sum,
                            float* __restrict__ seg_cnt, int n) {
    int i = blockIdx.x * blockDim.x + threadIdx.x;   // one thread per (node, 4 cols)
    int node = i >> 5;
    int lane = i & 31;
    if (node >= n) return;
    if (ntype[node] != 0) return;
    int g = batch[node];
    v4f v = *(const v4f*)(h + (size_t)node * HIDDEN + lane * 4);
    float* d = seg_sum + (size_t)g * HIDDEN + lane * 4;
    atomicAdd(d + 0, v.x);
    atomicAdd(d + 1, v.y);
    atomicAdd(d + 2, v.z);
    atomicAdd(d + 3, v.w);
    if (lane == 0) atomicAdd(&seg_cnt[g], 1.0f);
}

// ---------------------------------------------------------------- tiny head: (64,128)@(128,49)^T
__global__ void head_kernel(const float* __restrict__ seg_sum, const float* __restrict__ seg_cnt,
                            const float* __restrict__ lin_w, const float* __restrict__ lin_b,
                            float* __restrict__ out, int n_out) {
    int idx = blockIdx.x * blockDim.x + threadIdx.x;
    if (idx >= N_GRAPHS * n_out) return;
    int g = idx / n_out, j = idx % n_out;
    float inv = 1.0f / fmaxf(seg_cnt[g], 1.0f);
    float acc = lin_b[j];
    for (int c = 0; c < HIDDEN; ++c)
        acc += seg_sum[(size_t)g * HIDDEN + c] * inv * lin_w[(size_t)j * HIDDEN + c];
    out[idx] = acc;
}

// ---------------------------------------------------------------- launch
extern "C" void kernel_launch(void* const* d_in, const int* in_sizes, int n_in,
                              void* d_out, int out_size, void* d_ws, size_t ws_size,
                              hipStream_t stream) {
    const int*   x      = (const int*)d_in[0];
    const int*   eidx   = (const int*)d_in[1];
    const int*   etype  = (const int*)d_in[2];
    const int*   batch  = (const int*)d_in[3];
    const int*   ntype  = (const int*)d_in[4];
    const float* emb    = (const float*)d_in[6];
    const float* w_root = (const float*)d_in[7];
    const float* w_rel  = (const float*)d_in[8];
    const float* cbias  = (const float*)d_in[9];
    const float* lin_w  = (const float*)d_in[10];
    const float* lin_b  = (const float*)d_in[11];

    const int E = in_sizes[2];
    const int N = in_sizes[3];
    const int n_out = in_sizes[11];                  // 49
    const int* src = eidx;
    const int* dst = eidx + E;

    // workspace carve-up (~157 MB)
    char* ws = (char*)d_ws;
    float* feat    = (float*)ws; ws += (size_t)N * EMBED * 4;
    float* tmp     = (float*)ws; ws += (size_t)N * EMBED * 4;
    float* agg     = (float*)ws; ws += (size_t)N * HIDDEN * 4;
    int*   cnt     = (int*)  ws; ws += (size_t)N * NUM_RELS * 4;
    float* seg_sum = (float*)ws; ws += (size_t)N_GRAPHS * HIDDEN * 4;
    float* seg_cnt = (float*)ws; ws += (size_t)N_GRAPHS * 4;
    (void)ws_size; (void)n_in; (void)out_size;

    hipMemsetAsync(cnt, 0, (size_t)N * NUM_RELS * 4, stream);
    count_kernel<<<(E + 255) / 256, 256, 0, stream>>>(dst, etype, cnt, E);
    embed_kernel<<<(N * 32 + 255) / 256, 256, 0, stream>>>(x, emb, feat, N);

    const int gblocks = (N + 127) / 128;             // 128 rows per block (8 waves x 16)
    // agg = feat @ w_root
    gemm128_wmma<<<gblocks, 256, 0, stream>>>(feat, w_root, agg, nullptr, 0, nullptr, 0, N);

    for (int r = 0; r < NUM_RELS; ++r) {
        hipMemsetAsync(tmp, 0, (size_t)N * EMBED * 4, stream);
        scatter_kernel<<<(E + 7) / 8, 256, 0, stream>>>(src, dst, etype, feat, tmp, E, r);
        int mode = (r == NUM_RELS - 1) ? 2 : 1;      // final pass adds bias + relu
        gemm128_wmma<<<gblocks, 256, 0, stream>>>(
            tmp, w_rel + (size_t)r * EMBED * HIDDEN, agg, cnt, r, cbias, mode, N);
    }

    hipMemsetAsync(seg_sum, 0, ((size_t)N_GRAPHS * HIDDEN + N_GRAPHS) * 4, stream);
    pool_kernel<<<(N * 32 + 255) / 256, 256, 0, stream>>>(agg, batch, ntype, seg_sum, seg_cnt, N);
    head_kernel<<<(N_GRAPHS * n_out + 255) / 256, 256, 0, stream>>>(
        seg_sum, seg_cnt, lin_w, lin_b, (float*)d_out, n_out);
}